// NetRGCN_2439541424711
// MI455X (gfx1250) — compile-verified
//
#include <hip/hip_runtime.h>
#include <hip/hip_bf16.h>
#include <math.h>

typedef float v2f __attribute__((ext_vector_type(2)));
typedef float v8f __attribute__((ext_vector_type(8)));

#define NNODES 60000
#define DIM    300
#define NREL   5
#define NEDGE  1000000
#define NBATCH 32
#define NCLS   25000   // classes per head (global and sense are both 25000)

// workspace layout (float offsets)
#define WS_ACCUM 0
#define WS_CNT   (NBATCH * NREL * DIM)          // 48000
#define WS_W     (WS_CNT + NBATCH * NREL)       // 48160
#define WS_H     (WS_W + NREL * DIM * DIM)      // 498160

// ---------------------------------------------------------------------------
// zero a float region
__global__ void zero_kernel(float* __restrict__ p, int n) {
    int i = blockIdx.x * blockDim.x + threadIdx.x;
    if (i < n) p[i] = 0.0f;
}

// ---------------------------------------------------------------------------
// Scan all edges; for edges whose dst is one of the 32 batch nodes,
// scatter-add x[src] into accum[(b*R + rel)*D + :] and bump cnt[b*R + rel].
__global__ __launch_bounds__(256)
void edge_scatter_kernel(const int* __restrict__ edge_index,
                         const int* __restrict__ edge_type,
                         const int* __restrict__ cur,
                         const float* __restrict__ x,
                         float* __restrict__ accum,
                         float* __restrict__ cnt) {
    __shared__ int sc[NBATCH];
    if (threadIdx.x < NBATCH) sc[threadIdx.x] = cur[threadIdx.x];
    __syncthreads();
    int e = blockIdx.x * blockDim.x + threadIdx.x;
    if (e >= NEDGE) return;
    int src = edge_index[e];
    int dst = edge_index[NEDGE + e];
    int rel = edge_type[e];
    const float* xs = x + (size_t)src * DIM;
    #pragma unroll 1
    for (int b = 0; b < NBATCH; ++b) {
        if (sc[b] == dst) {
            float* acc = accum + (size_t)(b * NREL + rel) * DIM;
            atomicAdd(&cnt[b * NREL + rel], 1.0f);
            for (int i = 0; i < DIM; ++i) atomicAdd(&acc[i], xs[i]);
        }
    }
}

// ---------------------------------------------------------------------------
// W[r] = sum_b comp[r][b] * basis[b]   (R x D x D)
__global__ void relw_kernel(const float* __restrict__ comp,
                            const float* __restrict__ basis,
                            float* __restrict__ W) {
    int idx = blockIdx.x * blockDim.x + threadIdx.x;
    if (idx >= NREL * DIM * DIM) return;
    int r = idx / (DIM * DIM);
    int ij = idx - r * (DIM * DIM);
    float s = 0.0f;
    #pragma unroll
    for (int b = 0; b < NREL; ++b)
        s += comp[r * NREL + b] * basis[(size_t)b * DIM * DIM + ij];
    W[idx] = s;
}

// ---------------------------------------------------------------------------
// h[b][j] = relu( bias[j] + x[n]·root[:,j] + sum_r (accum_{b,r}/max(cnt,1))·W_r[:,j] )
__global__ __launch_bounds__(256)
void hidden_kernel(const float* __restrict__ x,
                   const int* __restrict__ cur,
                   const float* __restrict__ accum,
                   const float* __restrict__ cnt,
                   const float* __restrict__ W,
                   const float* __restrict__ root,
                   const float* __restrict__ bias,
                   float* __restrict__ h) {
    int idx = blockIdx.x * blockDim.x + threadIdx.x;
    if (idx >= NBATCH * DIM) return;
    int b = idx / DIM;
    int j = idx - b * DIM;
    int n = cur[b];
    float s = bias[j];
    const float* xn = x + (size_t)n * DIM;
    for (int i = 0; i < DIM; ++i)
        s += xn[i] * root[(size_t)i * DIM + j];          // coalesced across j
    for (int r = 0; r < NREL; ++r) {
        float c = cnt[b * NREL + r];
        float inv = 1.0f / fmaxf(c, 1.0f);
        const float* ar = accum + (size_t)(b * NREL + r) * DIM;
        const float* Wr = W + (size_t)r * DIM * DIM;
        float t = 0.0f;
        for (int i = 0; i < DIM; ++i)
            t += ar[i] * Wr[(size_t)i * DIM + j];        // coalesced across j
        s += t * inv;
    }
    h[(size_t)b * DIM + j] = fmaxf(s, 0.0f);
}

// ---------------------------------------------------------------------------
// logits[32 x 25000] = h[32 x 300] @ w[25000 x 300]^T + bias, via
// V_WMMA_F32_16X16X4_F32. One wave per 16-class tile, both 16-row M tiles.
// A (16x4 f32): lanes 0-15 hold K=k,k+1 for M=lane; lanes 16-31 hold K=k+2,k+3.
// B (4x16 f32): lanes 0-15 hold K=k,k+1 for N=lane; lanes 16-31 hold K=k+2,k+3.
__global__ __launch_bounds__(128)
void logits_wmma_kernel(const float* __restrict__ h,
                        const float* __restrict__ w,
                        const float* __restrict__ bvec,
                        float* __restrict__ out) {
    int lane = threadIdx.x & 31;
    int wv   = threadIdx.x >> 5;
    int ntiles = (NCLS + 15) / 16;              // 1563
    int tile = blockIdx.x * 4 + wv;
    if (tile >= ntiles) return;                 // wave-uniform exit
    int nloc = lane & 15;
    int hiHalf = lane >> 4;
    int koff = hiHalf ? 2 : 0;
    int col = tile * 16 + nloc;
    float mask = (col < NCLS) ? 1.0f : 0.0f;    // zero B frag for OOB columns
    int ccol = (col < NCLS) ? col : (NCLS - 1); // clamp pointer, no divergence
    const float* wrow = w + (size_t)ccol * DIM;
    const float* ha0 = h + (size_t)nloc * DIM;          // A rows 0..15
    const float* ha1 = ha0 + (size_t)16 * DIM;          // A rows 16..31

    v8f acc0 = {0.f, 0.f, 0.f, 0.f, 0.f, 0.f, 0.f, 0.f};
    v8f acc1 = {0.f, 0.f, 0.f, 0.f, 0.f, 0.f, 0.f, 0.f};

    for (int k = 0; k < DIM; k += 4) {          // 75 iters, K=300
        v2f a0 = *reinterpret_cast<const v2f*>(ha0 + k + koff);
        v2f a1 = *reinterpret_cast<const v2f*>(ha1 + k + koff);
        v2f bb = *reinterpret_cast<const v2f*>(wrow + k + koff);
        bb.x *= mask;
        bb.y *= mask;
        acc0 = __builtin_amdgcn_wmma_f32_16x16x4_f32(
            false, a0, false, bb, (short)0, acc0, false, false);
        acc1 = __builtin_amdgcn_wmma_f32_16x16x4_f32(
            false, a1, false, bb, (short)0, acc1, false, false);
    }

    if (col < NCLS) {
        float bv = bvec[col];
        int rbase = hiHalf ? 8 : 0;             // C/D: lanes 16-31 hold M=v+8
        #pragma unroll
        for (int v = 0; v < 8; ++v) {
            out[(size_t)(rbase + v) * NCLS + col]      = acc0[v] + bv;
            out[(size_t)(16 + rbase + v) * NCLS + col] = acc1[v] + bv;
        }
    }
}

// ---------------------------------------------------------------------------
// in-place log_softmax over each of the 64 rows (32 global + 32 sense)
__global__ __launch_bounds__(256)
void logsoftmax_kernel(float* __restrict__ out) {
    float* row = out + (size_t)blockIdx.x * NCLS;
    __shared__ float red[256];
    int tid = threadIdx.x;

    float m = -INFINITY;
    for (int i = tid; i < NCLS; i += 256) m = fmaxf(m, row[i]);
    red[tid] = m;
    __syncthreads();
    for (int s = 128; s > 0; s >>= 1) {
        if (tid < s) red[tid] = fmaxf(red[tid], red[tid + s]);
        __syncthreads();
    }
    m = red[0];
    __syncthreads();

    float sum = 0.0f;
    for (int i = tid; i < NCLS; i += 256) sum += expf(row[i] - m);
    red[tid] = sum;
    __syncthreads();
    for (int s = 128; s > 0; s >>= 1) {
        if (tid < s) red[tid] += red[tid + s];
        __syncthreads();
    }
    float lse = m + logf(red[0]);
    __syncthreads();

    for (int i = tid; i < NCLS; i += 256) row[i] = row[i] - lse;
}

// ---------------------------------------------------------------------------
extern "C" void kernel_launch(void* const* d_in, const int* in_sizes, int n_in,
                              void* d_out, int out_size, void* d_ws, size_t ws_size,
                              hipStream_t stream) {
    const float* x          = (const float*)d_in[0];
    const int*   edge_index = (const int*)  d_in[1];
    const int*   edge_type  = (const int*)  d_in[2];
    const int*   cur        = (const int*)  d_in[3];
    const float* basis      = (const float*)d_in[4];
    const float* comp       = (const float*)d_in[5];
    const float* root       = (const float*)d_in[6];
    const float* bias       = (const float*)d_in[7];
    const float* w_global   = (const float*)d_in[8];
    const float* b_global   = (const float*)d_in[9];
    const float* w_sense    = (const float*)d_in[10];
    const float* b_sense    = (const float*)d_in[11];
    float* out = (float*)d_out;
    float* ws  = (float*)d_ws;

    float* accum = ws + WS_ACCUM;   // 32*5*300
    float* cnt   = ws + WS_CNT;     // 32*5
    float* W     = ws + WS_W;       // 5*300*300
    float* h     = ws + WS_H;       // 32*300

    int nz = NBATCH * NREL * DIM + NBATCH * NREL;
    zero_kernel<<<(nz + 255) / 256, 256, 0, stream>>>(accum, nz);

    edge_scatter_kernel<<<(NEDGE + 255) / 256, 256, 0, stream>>>(
        edge_index, edge_type, cur, x, accum, cnt);

    int nw = NREL * DIM * DIM;
    relw_kernel<<<(nw + 255) / 256, 256, 0, stream>>>(comp, basis, W);

    hidden_kernel<<<(NBATCH * DIM + 255) / 256, 256, 0, stream>>>(
        x, cur, accum, cnt, W, root, bias, h);

    int ntiles = (NCLS + 15) / 16;              // 1563
    int nblk = (ntiles + 3) / 4;                // 4 waves (tiles) per block
    logits_wmma_kernel<<<nblk, 128, 0, stream>>>(h, w_global, b_global, out);
    logits_wmma_kernel<<<nblk, 128, 0, stream>>>(h, w_sense, b_sense,
                                                 out + (size_t)NBATCH * NCLS);

    logsoftmax_kernel<<<64, 256, 0, stream>>>(out);
}